// ARMLoss_21973052686930
// MI455X (gfx1250) — compile-verified
//
#include <hip/hip_runtime.h>
#include <stdint.h>

#define BB 64
#define PP 32768
#define OO 16
#define TILE 2048
#define TPB 256
#define PPT 8   // priors per thread (TILE = TPB*PPT)

typedef float v2f __attribute__((ext_vector_type(2)));
typedef float v8f __attribute__((ext_vector_type(8)));
typedef unsigned long long u64;

// ---- exact f32 wave(32)-sum on the matrix pipe: V_WMMA_F32_16X16X4_F32 ----
// A(16x4): lane L<16 holds {K0=v[L], K1=0}, lane L>=16 holds {K2=v[L], K3=0}
// B(4x16)=ones  =>  D[m][*] = v[m] + v[m+16].
// Sum of the 8 D VGPRs in a lane covers rows 0-7 (lanes<16) / 8-15 (lanes>=16);
// one xor-16 shuffle-add yields the full 32-lane sum (f32, RNE, exact order-free).
__device__ __forceinline__ float wave_sum_wmma(float v) {
  v2f a; a[0] = v;    a[1] = 0.0f;
  v2f o; o[0] = 1.0f; o[1] = 1.0f;
  v8f c = {};
  v8f d = __builtin_amdgcn_wmma_f32_16x16x4_f32(false, a, false, o, (short)0, c,
                                                false, false);
  float s = d[0] + d[1] + d[2] + d[3] + d[4] + d[5] + d[6] + d[7];
  s += __shfl_xor(s, 16, 32);
  return s;
}

__device__ __forceinline__ float smoothl1(float d) {
  float a = fabsf(d);
  return (a < 1.0f) ? 0.5f * d * d : a - 0.5f;
}

// ================= Phase A: per-truth best prior (argmax over P) =============
// Packed key: (float_bits(iou) << 32) | ~prior_idx  -> max == best iou,
// ties resolved to LOWEST prior index (matches jnp.argmax first-occurrence).
__global__ __launch_bounds__(TPB) void phaseA_k(const float* __restrict__ priors,
                                                const float* __restrict__ targets,
                                                u64* __restrict__ bestP) {
  __shared__ float spri[TILE * 4];           // 32 KB prior tile
  __shared__ float str[OO * 4];
  __shared__ float sta[OO];
  __shared__ u64   sred[OO];
  const int t  = threadIdx.x;
  const int b  = blockIdx.y;
  const int p0 = blockIdx.x * TILE;

  // ---- async-stage the prior tile into LDS (CDNA5 ASYNC path) ----
  unsigned ldsbase = (unsigned)(uintptr_t)(void*)spri;  // low 32 bits = LDS offset
  const float* gp = priors + (size_t)p0 * 4;
#pragma unroll
  for (int i = 0; i < PPT; ++i) {
    unsigned voff  = (unsigned)((i * TPB + t) * 16);
    unsigned laddr = ldsbase + voff;
    asm volatile("global_load_async_to_lds_b128 %0, %1, %2 offset:0"
                 :: "v"(laddr), "v"(voff), "s"(gp) : "memory");
  }
  if (t < OO) {
    const float* tp = targets + ((size_t)b * OO + t) * 5;
    float x0 = tp[0], y0 = tp[1], x1 = tp[2], y1 = tp[3];
    str[t * 4 + 0] = x0; str[t * 4 + 1] = y0;
    str[t * 4 + 2] = x1; str[t * 4 + 3] = y1;
    sta[t] = (x1 - x0) * (y1 - y0);
    sred[t] = 0ull;
  }
  asm volatile("s_wait_asynccnt 0x0" ::: "memory");
  __syncthreads();

  u64 best[OO];
#pragma unroll
  for (int o = 0; o < OO; ++o) best[o] = 0ull;

  for (int i = 0; i < PPT; ++i) {
    int pi = i * TPB + t;
    int p  = p0 + pi;
    float cx = spri[pi * 4 + 0], cy = spri[pi * 4 + 1];
    float w  = spri[pi * 4 + 2], h  = spri[pi * 4 + 3];
    float px0 = cx - 0.5f * w, py0 = cy - 0.5f * h;
    float px1 = cx + 0.5f * w, py1 = cy + 0.5f * h;
    float pa  = (px1 - px0) * (py1 - py0);
    u64 lo = (u64)(unsigned)(~(unsigned)p);
#pragma unroll
    for (int o = 0; o < OO; ++o) {
      float ix0 = fmaxf(str[o * 4 + 0], px0), iy0 = fmaxf(str[o * 4 + 1], py0);
      float ix1 = fminf(str[o * 4 + 2], px1), iy1 = fminf(str[o * 4 + 3], py1);
      float iw = fmaxf(ix1 - ix0, 0.0f), ih = fmaxf(iy1 - iy0, 0.0f);
      float inter = iw * ih;
      float v = inter / (sta[o] + pa - inter);
      u64 pk = ((u64)__float_as_uint(v) << 32) | lo;
      if (pk > best[o]) best[o] = pk;
    }
  }
#pragma unroll
  for (int o = 0; o < OO; ++o) atomicMax(&sred[o], best[o]);
  __syncthreads();
  if (t < OO) atomicMax(&bestP[(size_t)b * OO + t], sred[t]);
}

// ====== Phase B: per-prior match, CE, smooth-L1, rank array, L0 histogram ====
__global__ __launch_bounds__(TPB) void phaseB_k(const float* __restrict__ loc,
                                                const float* __restrict__ conf,
                                                const float* __restrict__ priors,
                                                const float* __restrict__ targets,
                                                const u64* __restrict__ bestP,
                                                float* __restrict__ rank,
                                                unsigned* __restrict__ ghist,
                                                float* __restrict__ numpos,
                                                float* __restrict__ sumcepos,
                                                float* __restrict__ losslG) {
  __shared__ float    str[OO * 4];
  __shared__ float    sta[OO];
  __shared__ int      sbp[OO];
  __shared__ unsigned shist[256];
  const int t  = threadIdx.x;
  const int b  = blockIdx.y;
  const int p0 = blockIdx.x * TILE;

  shist[t] = 0u;
  if (t < OO) {
    const float* tp = targets + ((size_t)b * OO + t) * 5;
    float x0 = tp[0], y0 = tp[1], x1 = tp[2], y1 = tp[3];
    str[t * 4 + 0] = x0; str[t * 4 + 1] = y0;
    str[t * 4 + 2] = x1; str[t * 4 + 3] = y1;
    sta[t] = (x1 - x0) * (y1 - y0);
    sbp[t] = (int)(~(unsigned)(bestP[(size_t)b * OO + t] & 0xffffffffull));
  }
  __syncthreads();

  float acc_l = 0.0f, acc_ce = 0.0f, acc_np = 0.0f;

  for (int i = 0; i < PPT; ++i) {
    int p = p0 + i * TPB + t;
    float4 pr = ((const float4*)priors)[p];
    float px0 = pr.x - 0.5f * pr.z, py0 = pr.y - 0.5f * pr.w;
    float px1 = pr.x + 0.5f * pr.z, py1 = pr.y + 0.5f * pr.w;
    float pa  = (px1 - px0) * (py1 - py0);

    float bv = -1.0f; int bi = 0;
#pragma unroll
    for (int o = 0; o < OO; ++o) {
      float ix0 = fmaxf(str[o * 4 + 0], px0), iy0 = fmaxf(str[o * 4 + 1], py0);
      float ix1 = fminf(str[o * 4 + 2], px1), iy1 = fminf(str[o * 4 + 3], py1);
      float iw = fmaxf(ix1 - ix0, 0.0f), ih = fmaxf(iy1 - iy0, 0.0f);
      float inter = iw * ih;
      float v = inter / (sta[o] + pa - inter);
      if (v > bv) { bv = v; bi = o; }     // strict > : first-max (argmax semantics)
    }
#pragma unroll
    for (int o = 0; o < OO; ++o)          // forced matches (scatter, last write wins)
      if (sbp[o] == p) { bv = 2.0f; bi = o; }

    bool pos = (bv >= 0.5f);

    float2 cf = ((const float2*)conf)[(size_t)b * PP + p];
    float m   = fmaxf(cf.x, cf.y);
    float lse = m + logf(expf(cf.x - m) + expf(cf.y - m));
    float ce  = lse - (pos ? cf.y : cf.x);           // ce >= 0 always
    float rk  = pos ? 0.0f : ce;
    rank[(size_t)b * PP + p] = rk;
    atomicAdd(&shist[__float_as_uint(rk) >> 24], 1u);

    if (pos) {
      acc_ce += ce;
      acc_np += 1.0f;
      float mnx = str[bi * 4 + 0], mny = str[bi * 4 + 1];
      float mxx = str[bi * 4 + 2], mxy = str[bi * 4 + 3];
      float4 ld = ((const float4*)loc)[(size_t)b * PP + p];
      float gx = ((mnx + mxx) * 0.5f - pr.x) / (0.1f * pr.z);
      float gy = ((mny + mxy) * 0.5f - pr.y) / (0.1f * pr.w);
      float gw = logf((mxx - mnx) / pr.z) / 0.2f;
      float gh = logf((mxy - mny) / pr.w) / 0.2f;
      acc_l += smoothl1(ld.x - gx) + smoothl1(ld.y - gy) +
               smoothl1(ld.z - gw) + smoothl1(ld.w - gh);
    }
  }

  // wave reductions on the matrix pipe (all lanes active => EXEC all-ones)
  float wl = wave_sum_wmma(acc_l);
  float wc = wave_sum_wmma(acc_ce);
  float wn = wave_sum_wmma(acc_np);
  if ((t & 31) == 0) {
    atomicAdd(losslG, wl);
    atomicAdd(&sumcepos[b], wc);
    atomicAdd(&numpos[b], wn);
  }
  __syncthreads();
  atomicAdd(&ghist[b * 256 + t], shist[t]);
}

// === radix-select scan: per-batch, pick byte bin containing k-th largest =====
__global__ void scan_k(unsigned* __restrict__ ghist,
                       const float* __restrict__ numpos,
                       unsigned* __restrict__ prefix,
                       unsigned* __restrict__ krem, int level) {
  int b = threadIdx.x;
  if (b >= BB) return;
  unsigned k;
  if (level == 0) {
    int np = (int)(numpos[b] + 0.5f);
    long long kk = 3LL * np;
    if (kk > PP - 1) kk = PP - 1;
    k = (unsigned)kk;                 // num_neg = min(3*num_pos, P-1)
    prefix[b] = 0u;
  } else {
    k = krem[b];
  }
  unsigned* h = ghist + b * 256;
  unsigned cum = 0; int bin = 0;
  for (int i = 255; i >= 0; --i) {
    unsigned c = h[i];
    if (cum + c >= k) { bin = i; k -= cum; break; }
    cum += c;
  }
  for (int i = 0; i < 256; ++i) h[i] = 0;  // reset for next level
  prefix[b] = (prefix[b] << 8) | (unsigned)bin;
  krem[b]   = k;
}

// === refine histogram on next byte for elements matching the current prefix ==
__global__ __launch_bounds__(TPB) void hist_k(const float* __restrict__ rank,
                                              unsigned* __restrict__ ghist,
                                              const unsigned* __restrict__ prefix,
                                              int level) {
  __shared__ unsigned shist[256];
  const int t  = threadIdx.x;
  const int b  = blockIdx.y;
  const int p0 = blockIdx.x * TILE;
  shist[t] = 0u;
  __syncthreads();
  unsigned pref = prefix[b];
  int shc = 32 - 8 * level;   // compare bits[31 : 32-8*level]
  int shb = 24 - 8 * level;   // next byte
  for (int i = 0; i < PPT; ++i) {
    int p = p0 + i * TPB + t;
    unsigned bits = __float_as_uint(rank[(size_t)b * PP + p]);
    if ((bits >> shc) == pref)
      atomicAdd(&shist[(bits >> shb) & 0xFFu], 1u);
  }
  __syncthreads();
  atomicAdd(&ghist[b * 256 + t], shist[t]);
}

// === sum of rank values strictly above the exact 32-bit threshold ============
__global__ __launch_bounds__(TPB) void finsum_k(const float* __restrict__ rank,
                                                const unsigned* __restrict__ prefix,
                                                float* __restrict__ sumgt) {
  const int t  = threadIdx.x;
  const int b  = blockIdx.y;
  const int p0 = blockIdx.x * TILE;
  unsigned T = prefix[b];
  float s = 0.0f;
  for (int i = 0; i < PPT; ++i) {
    int p = p0 + i * TPB + t;
    unsigned bits = __float_as_uint(rank[(size_t)b * PP + p]);
    if (bits > T) s += __uint_as_float(bits);   // monotone: values >= 0
  }
  float w = wave_sum_wmma(s);
  if ((t & 31) == 0) atomicAdd(&sumgt[b], w);
}

// === final combine: loss_c per batch = sum_gt + r*T + sum_ce_pos =============
__global__ void combine_k(const float* __restrict__ sumgt,
                          const unsigned* __restrict__ prefix,
                          const unsigned* __restrict__ krem,
                          const float* __restrict__ sumcepos,
                          const float* __restrict__ numpos,
                          const float* __restrict__ losslG,
                          float* __restrict__ out) {
  __shared__ float sc[BB];
  __shared__ float sn[BB];
  int b = threadIdx.x;
  if (b < BB) {
    float lc = sumgt[b] + (float)krem[b] * __uint_as_float(prefix[b]) + sumcepos[b];
    sc[b] = lc;
    sn[b] = numpos[b];
  }
  __syncthreads();
  if (b == 0) {
    float C = 0.0f, N = 0.0f;
    for (int i = 0; i < BB; ++i) { C += sc[i]; N += sn[i]; }
    out[0] = losslG[0] / N;
    out[1] = C / N;
  }
}

extern "C" void kernel_launch(void* const* d_in, const int* in_sizes, int n_in,
                              void* d_out, int out_size, void* d_ws, size_t ws_size,
                              hipStream_t stream) {
  const float* loc     = (const float*)d_in[0];   // (64, 32768, 4)
  const float* conf    = (const float*)d_in[1];   // (64, 32768, 2)
  const float* priors  = (const float*)d_in[2];   // (32768, 4)
  const float* targets = (const float*)d_in[3];   // (64, 16, 5)
  float* out = (float*)d_out;

  // ---- workspace layout (~8.46 MB) ----
  char* w = (char*)d_ws;
  size_t off = 0;
  float* rank = (float*)(w + off);          off += (size_t)BB * PP * 4;   // 8 MB
  size_t meta0 = off;
  u64*      bestP    = (u64*)(w + off);     off += (size_t)BB * OO * 8;
  unsigned* ghist    = (unsigned*)(w + off);off += (size_t)BB * 256 * 4;
  float*    numpos   = (float*)(w + off);   off += BB * 4;
  float*    sumcepos = (float*)(w + off);   off += BB * 4;
  float*    losslG   = (float*)(w + off);   off += 64;   // padded slot
  unsigned* prefix   = (unsigned*)(w + off);off += BB * 4;
  unsigned* krem     = (unsigned*)(w + off);off += BB * 4;
  float*    sumgt    = (float*)(w + off);   off += BB * 4;

  hipMemsetAsync(w + meta0, 0, off - meta0, stream);  // zero all accumulators

  dim3 grid(PP / TILE, BB);   // (16, 64)

  phaseA_k<<<grid, TPB, 0, stream>>>(priors, targets, bestP);
  phaseB_k<<<grid, TPB, 0, stream>>>(loc, conf, priors, targets, bestP,
                                     rank, ghist, numpos, sumcepos, losslG);
  scan_k<<<1, BB, 0, stream>>>(ghist, numpos, prefix, krem, 0);
  for (int lvl = 1; lvl <= 3; ++lvl) {
    hist_k<<<grid, TPB, 0, stream>>>(rank, ghist, prefix, lvl);
    scan_k<<<1, BB, 0, stream>>>(ghist, numpos, prefix, krem, lvl);
  }
  finsum_k<<<grid, TPB, 0, stream>>>(rank, prefix, sumgt);
  combine_k<<<1, BB, 0, stream>>>(sumgt, prefix, krem, sumcepos, numpos,
                                  losslG, out);
}